// Model_10007273799960
// MI455X (gfx1250) — compile-verified
//
#include <hip/hip_runtime.h>
#include <hip/hip_bf16.h>
#include <stdint.h>

#define DIN 256
#define DOUT 128

typedef float v2f __attribute__((ext_vector_type(2)));
typedef float v8f __attribute__((ext_vector_type(8)));
typedef unsigned int v4u __attribute__((ext_vector_type(4)));
typedef int v4i __attribute__((ext_vector_type(4)));
typedef int v8i __attribute__((ext_vector_type(8)));

#ifndef __has_builtin
#define __has_builtin(x) 0
#endif

#if __has_builtin(__builtin_amdgcn_tensor_load_to_lds) && __has_builtin(__builtin_amdgcn_s_wait_tensorcnt)
#define USE_TDM 1
#else
#define USE_TDM 0
#endif

// ---------------- degree / norm ----------------

__global__ void deg_init_kernel(float* __restrict__ deg, int n) {
    int i = blockIdx.x * blockDim.x + threadIdx.x;
    if (i < n) deg[i] = 1.0f;  // self-loop contributes 1
}

__global__ void deg_acc_kernel(const long long* __restrict__ dst,
                               float* __restrict__ deg, int n_edges) {
    int e = blockIdx.x * blockDim.x + threadIdx.x;
    if (e < n_edges) {
        long long d = dst[e];
        unsafeAtomicAdd(&deg[d], 1.0f);   // native global_atomic_add_f32
    }
}

__global__ void dinv_kernel(const float* __restrict__ deg,
                            float* __restrict__ dinv, int n) {
    int i = blockIdx.x * blockDim.x + threadIdx.x;
    if (i < n) dinv[i] = rsqrtf(deg[i]);
}

// ---------------- GEMM h = x @ W via fp32 WMMA ----------------
// One block = 16-row strip of x staged in LDS (via Tensor Data Mover when
// available, with TDM pad feature reproducing the 260-float stride);
// 8 waves each own a 16-col tile. 64 x V_WMMA_F32_16X16X4_F32 per wave.

#define LDS_STRIDE 260  // 256 + 4 pad DWORDs: rows land on distinct LDS banks

__global__ __launch_bounds__(256) void gemm_kernel(const float* __restrict__ x,
                                                   const float* __restrict__ W,
                                                   float* __restrict__ h) {
    __shared__ float As[16 * LDS_STRIDE];

    const int tid  = threadIdx.x;
    const int wave = tid >> 5;
    const int lane = tid & 31;
    const int m0   = blockIdx.x * 16;

#if USE_TDM
    // One TDM descriptor DMAs the whole 16x256 fp32 strip into LDS,
    // padding +4 DWORDs after every 256 DWORDs (row) => stride 260.
    if (tid < 32) {  // wave 0 issues; TDM ignores EXEC, one op per wave
        const uint64_t ga = (uint64_t)(uintptr_t)(x + (size_t)m0 * DIN);
        const uint32_t lds_base = (uint32_t)(uintptr_t)(void*)As;
        v4u g0;
        g0[0] = 1u;                                  // count=1, user mode
        g0[1] = lds_base;                            // lds_addr (bytes)
        g0[2] = (uint32_t)ga;                        // global_addr[31:0]
        g0[3] = ((uint32_t)(ga >> 32) & 0x01FFFFFFu) // global_addr[56:32]
                | 0x80000000u;                       // type=2 ("image")
        v8i g1;
        g1[0] = (int)((2u << 16)        // data_size = 4 bytes
                    | (1u << 20)        // pad_enable
                    | (7u << 22)        // pad_interval: every 256 DWORDs
                    | (3u << 25));      // pad_amount: 4 DWORDs
        g1[1] = (int)(256u << 16);      // tensor_dim0 = 256 (bits 79:48 lo16)
        g1[2] = (int)(16u << 16);       // tensor_dim1 = 16  (bits 111:80 lo16)
        g1[3] = (int)(256u << 16);      // tile_dim0 = 256   (bits 127:112)
        g1[4] = 16;                     // tile_dim1 = 16    (bits 143:128)
        g1[5] = 256;                    // tensor_dim0_stride = 256 (lo32)
        g1[6] = (int)(256u << 16);      // tensor_dim1_stride lo16 (bits 223:208)
        g1[7] = 0;
        v4i z4 = {0, 0, 0, 0};          // 2D tile: groups 2/3 unused
#if defined(__clang_major__) && (__clang_major__ >= 23)
        v8i z8 = {0, 0, 0, 0, 0, 0, 0, 0};
        __builtin_amdgcn_tensor_load_to_lds(g0, g1, z4, z4, z8, 0);
#else
        __builtin_amdgcn_tensor_load_to_lds(g0, g1, z4, z4, 0);
#endif
        __builtin_amdgcn_s_wait_tensorcnt(0);
    }
    __syncthreads();
#else
    // Fallback: cooperative coalesced stage of rows m0..m0+15.
    {
        const int row = tid >> 4;          // 0..15
        const int cb  = (tid & 15) * 16;   // 16 floats per thread
        const float4* src4 = (const float4*)(x + (size_t)(m0 + row) * DIN + cb);
        float* dstp = As + row * LDS_STRIDE + cb;
        #pragma unroll
        for (int j = 0; j < 4; ++j) {
            float4 v = src4[j];
            dstp[j * 4 + 0] = v.x;
            dstp[j * 4 + 1] = v.y;
            dstp[j * 4 + 2] = v.z;
            dstp[j * 4 + 3] = v.w;
        }
    }
    __syncthreads();
#endif

    const int n0 = wave * 16;       // this wave's N tile
    const int lm = lane & 15;       // M row (A) / N col (B,C,D)
    const int hi = lane >> 4;       // half-wave select
    const int n  = n0 + lm;

    const float* arow = As + lm * LDS_STRIDE;

    v8f c = {};
    #pragma unroll
    for (int k0 = 0; k0 < DIN; k0 += 4) {
        // A 16x4 f32 layout: lanes 0-15 -> K=k0,k0+1 ; lanes 16-31 -> K=k0+2,k0+3
        const int ka = k0 + hi * 2;
        v2f a;
        a.x = arow[ka];
        a.y = arow[ka + 1];
        // B 4x16 f32 layout (mirror): VGPR0 = row ka, VGPR1 = row ka+1, N = lane%16
        v2f bb;
        bb.x = W[ka * DOUT + n];
        bb.y = W[(ka + 1) * DOUT + n];
        c = __builtin_amdgcn_wmma_f32_16x16x4_f32(
                /*neg_a=*/false, a, /*neg_b=*/false, bb,
                /*c_mod=*/(short)0, c, /*reuse_a=*/false, /*reuse_b=*/false);
    }

    // C/D 16x16 f32 layout: VGPR v -> M = v + 8*hi, N = lane%16
    #pragma unroll
    for (int v = 0; v < 8; ++v) {
        h[(size_t)(m0 + v + hi * 8) * DOUT + n] = c[v];
    }
}

// ---------------- self-loop init: out = h * dinv^2 ----------------

__global__ void self_kernel(const float* __restrict__ h,
                            const float* __restrict__ dinv,
                            float* __restrict__ out, int total) {
    int idx = blockIdx.x * blockDim.x + threadIdx.x;
    if (idx >= total) return;
    int i = idx >> 7;  // node index (DOUT=128)
    float di = dinv[i];
    out[idx] = h[idx] * (di * di);
}

// ---------------- edge scatter: 1 wave per edge, float4 per lane ----------------

__global__ __launch_bounds__(256) void edge_kernel(const long long* __restrict__ src,
                                                   const long long* __restrict__ dst,
                                                   const float* __restrict__ dinv,
                                                   const float* __restrict__ h,
                                                   float* __restrict__ out,
                                                   int n_edges) {
    int e = blockIdx.x * 8 + (threadIdx.x >> 5);
    if (e >= n_edges) return;
    int lane = threadIdx.x & 31;

    long long s = src[e];
    long long d = dst[e];
    float norm = dinv[s] * dinv[d];

    const float4 hv = *(const float4*)(h + (size_t)s * DOUT + lane * 4);
    float* op = out + (size_t)d * DOUT + lane * 4;
    unsafeAtomicAdd(op + 0, hv.x * norm);
    unsafeAtomicAdd(op + 1, hv.y * norm);
    unsafeAtomicAdd(op + 2, hv.z * norm);
    unsafeAtomicAdd(op + 3, hv.w * norm);
}

// ---------------- bias + PReLU ----------------

__global__ void finish_kernel(float* __restrict__ out,
                              const float* __restrict__ b,
                              const float* __restrict__ a, int total) {
    int idx = blockIdx.x * blockDim.x + threadIdx.x;
    if (idx >= total) return;
    int ch = idx & (DOUT - 1);
    float v = out[idx] + b[ch];
    out[idx] = v > 0.0f ? v : a[ch] * v;
}

// ---------------- launcher ----------------

extern "C" void kernel_launch(void* const* d_in, const int* in_sizes, int n_in,
                              void* d_out, int out_size, void* d_ws, size_t ws_size,
                              hipStream_t stream) {
    const float*     x  = (const float*)d_in[0];
    const long long* ei = (const long long*)d_in[1];   // int64 edge_index [2, E]
    const float*     W  = (const float*)d_in[2];
    const float*     b  = (const float*)d_in[3];
    const float*     pa = (const float*)d_in[4];
    float* out = (float*)d_out;

    const int n_nodes = in_sizes[0] / DIN;   // 50000
    const int n_edges = in_sizes[1] / 2;     // 800000
    const long long* src = ei;
    const long long* dst = ei + n_edges;

    // workspace layout (floats): deg | dinv | h
    float* ws = (float*)d_ws;
    const int NPAD = 50048;  // 50000 rounded up, keeps 16B alignment
    float* deg  = ws;
    float* dinv = ws + NPAD;
    float* h    = ws + 2 * (size_t)NPAD;

    const int total = n_nodes * DOUT;

    deg_init_kernel<<<(n_nodes + 255) / 256, 256, 0, stream>>>(deg, n_nodes);
    deg_acc_kernel<<<(n_edges + 255) / 256, 256, 0, stream>>>(dst, deg, n_edges);
    dinv_kernel<<<(n_nodes + 255) / 256, 256, 0, stream>>>(deg, dinv, n_nodes);

    gemm_kernel<<<n_nodes / 16, 256, 0, stream>>>(x, W, h);   // 50000 = 3125*16 exactly

    self_kernel<<<(total + 255) / 256, 256, 0, stream>>>(h, dinv, out, total);
    edge_kernel<<<(n_edges + 7) / 8, 256, 0, stream>>>(src, dst, dinv, h, out, n_edges);
    finish_kernel<<<(total + 255) / 256, 256, 0, stream>>>(out, b, pa, total);
}